// CVRNNLayer_70489003262190
// MI455X (gfx1250) — compile-verified
//
#include <hip/hip_runtime.h>
#include <math.h>

typedef __attribute__((ext_vector_type(2))) float v2f;
typedef __attribute__((ext_vector_type(8))) float v8f;

#define NN     4096
#define BATCH  4
#define NT     256
#define TILES  (NN / 16)        // 256 row tiles
#define KPAIRS (NN / 2)         // 2048 WMMA iterations (K=4 each, doubled-K = 8192)
#define WAVES  4
#define KITER  (KPAIRS / WAVES) // 512 iterations per wave

// One time step: xcur = i*diag(omega)*xprev + B*xprev   (batched complex matvec)
// Complex folded into real GEMM with doubled K via the [[Br,-Bi],[Bi,Br]] embedding.
// Per WMMA i the K=4 slice is {Br[n0], Br[n1], Bi[n0], Bi[n1]}, n0=2i, n1=2i+1:
//   A operand (16x4): lanes 0-15 hold contiguous Br[m][2i..2i+1] (one b64 load),
//                     lanes 16-31 hold contiguous Bi[m][2i..2i+1]
//   B operand (4x16): rows K0,K1 = {xr[n][b], xi[n][b]}   (cols 0-3 re, 4-7 im)
//                     rows K2,K3 = {-xi[n][b], xr[n][b]}
__global__ __launch_bounds__(128)
void cvrnn_step(const float* __restrict__ Br, const float* __restrict__ Bi,
                const float* __restrict__ omega,
                const float* __restrict__ xprev,   // (b*NN+n)*2 + {re,im}
                float* __restrict__ xcur) {
    __shared__ float xpair[NN * 8];   // [n][0..3]=xr[n][b], [n][4..7]=xi[n][b]  (128 KB)
    __shared__ float accC[8 * 16];    // [col][row] partial C reduction

    const int tid = threadIdx.x;
    const int m0  = blockIdx.x * 16;

    // Stage the full state vector into LDS (coalesced b64 reads of (re,im) pairs).
    for (int k = tid; k < BATCH * NN; k += 128) {
        const int b = k >> 12;          // k / 4096
        const int n = k & (NN - 1);
        const float2 v = *(const float2*)(xprev + ((size_t)(b * NN + n) << 1));
        xpair[n * 8 + b]     = v.x;
        xpair[n * 8 + 4 + b] = v.y;
    }
    accC[tid] = 0.0f;                   // 128 threads zero 128 floats
    __syncthreads();

    const int lane = tid & 31;
    const int wave = tid >> 5;
    const int col  = lane & 15;         // output column / A row (m = m0+col)
    const int half = lane >> 4;         // 0 -> Br half (K0,K1), 1 -> Bi half (K2,K3)
    // B-operand element select per lane:
    //   half=0: pattern0[n][col] = xpair[n*8 + (col&7)]                (sign +)
    //   half=1: pattern1[n][col] = +/- xpair[n*8 + ((col+4)&7)]       (- for col%8<4)
    const int   sel = half ? ((col + 4) & 7) : (col & 7);
    const float s   = (half && ((col & 7) < 4)) ? -1.0f : 1.0f;

    const float2* pA =
        (const float2*)((half ? Bi : Br) + (size_t)(m0 + col) * NN);  // pA[i] = [2i..2i+1]
    const int iBase = wave * KITER;     // K split across the 4 waves

    v8f acc0 = {};
    v8f acc1 = {};
    #pragma unroll 4
    for (int ii = 0; ii < KITER; ii += 2) {
        const int i0 = iBase + ii;
        {
            const float2 av = pA[i0];
            v2f a  = { av.x, av.y };
            v2f bb = { s * xpair[16 * i0 + sel],
                       s * xpair[16 * i0 + 8 + sel] };
            acc0 = __builtin_amdgcn_wmma_f32_16x16x4_f32(
                       false, a, false, bb, (short)0, acc0, false, false);
        }
        const int i1 = i0 + 1;
        {
            const float2 av = pA[i1];
            v2f a  = { av.x, av.y };
            v2f bb = { s * xpair[16 * i1 + sel],
                       s * xpair[16 * i1 + 8 + sel] };
            acc1 = __builtin_amdgcn_wmma_f32_16x16x4_f32(
                       false, a, false, bb, (short)0, acc1, false, false);
        }
    }
    const v8f acc = acc0 + acc1;

    // C layout: lane holds column N=lane%16; VGPR v holds row M = v + 8*(lane/16).
    if (col < 8) {
        #pragma unroll
        for (int v = 0; v < 8; ++v) {
            atomicAdd(&accC[col * 16 + v + 8 * half], acc[v]);
        }
    }
    __syncthreads();

    // 128 threads finalize: add i*omega*x diagonal term, write interleaved complex.
    {
        const int r = tid >> 3;        // row within tile 0..15
        const int sc = tid & 7;        // column 0..7
        const int c = sc >> 2;         // 0 = real, 1 = imag
        const int b = sc & 3;          // batch
        const int m = m0 + r;
        float val = accC[sc * 16 + r];
        const float om = omega[b * NN + m];
        if (c == 0) val -= om * xpair[m * 8 + 4 + b];   // re += -omega * xi
        else        val += om * xpair[m * 8 + b];       // im +=  omega * xr
        xcur[(((size_t)(b * NN + m)) << 1) + c] = val;
    }
}

// t = 0: x0 = exp(i*theta) -> (cos, sin), written as interleaved complex.
__global__ void cvrnn_init(const float* __restrict__ angles, float* __restrict__ out0) {
    const int k = blockIdx.x * blockDim.x + threadIdx.x;   // k = b*NN + n
    if (k < BATCH * NN) {
        const float th = angles[k];
        out0[2 * k + 0] = cosf(th);
        out0[2 * k + 1] = sinf(th);
    }
}

extern "C" void kernel_launch(void* const* d_in, const int* in_sizes, int n_in,
                              void* d_out, int out_size, void* d_ws, size_t ws_size,
                              hipStream_t stream) {
    const float* Br     = (const float*)d_in[0];
    const float* Bi     = (const float*)d_in[1];
    const float* omega  = (const float*)d_in[2];
    const float* angles = (const float*)d_in[3];
    float* out = (float*)d_out;

    cvrnn_init<<<(BATCH * NN + 255) / 256, 256, 0, stream>>>(angles, out);

    for (int t = 1; t < NT; ++t) {
        const float* xprev = out + (size_t)(t - 1) * BATCH * NN * 2;
        float*       xcur  = out + (size_t)t * BATCH * NN * 2;
        cvrnn_step<<<TILES, 128, 0, stream>>>(Br, Bi, omega, xprev, xcur);
    }
}